// GCN_Dense_Aux_5609227288948
// MI455X (gfx1250) — compile-verified
//
#include <hip/hip_runtime.h>
#include <math.h>

// Problem constants (from reference)
#define N_NODES 20000
#define E_EDGES 320000
#define D_AUX   3
#define IN_F    300
#define H_F     256
#define OUT_F   256
#define NEG_SLOPE 0.2f
#define L2_EPS  1e-12f

typedef __attribute__((ext_vector_type(2))) float v2f;
typedef __attribute__((ext_vector_type(4))) float v4f;
typedef __attribute__((ext_vector_type(8))) float v8f;

// ---------------------------------------------------------------------------
// Kernel 1: per-edge mixing weights  v[e] = softmax(att) . vals[:, e]
// ---------------------------------------------------------------------------
__global__ __launch_bounds__(256) void edge_weights_kernel(
    const float* __restrict__ a_vals, const float* __restrict__ r_vals,
    const float* __restrict__ a_att,  const float* __restrict__ r_att,
    float* __restrict__ v1, float* __restrict__ v2) {
  int e = blockIdx.x * blockDim.x + threadIdx.x;
  if (e >= E_EDGES) return;

  float a0 = a_att[0], a1 = a_att[1], a2 = a_att[2];
  float ma = fmaxf(a0, fmaxf(a1, a2));
  float ea0 = __expf(a0 - ma), ea1 = __expf(a1 - ma), ea2 = __expf(a2 - ma);
  float ia = 1.0f / (ea0 + ea1 + ea2);

  float r0 = r_att[0], r1 = r_att[1], r2 = r_att[2];
  float mr = fmaxf(r0, fmaxf(r1, r2));
  float er0 = __expf(r0 - mr), er1 = __expf(r1 - mr), er2 = __expf(r2 - mr);
  float ir = 1.0f / (er0 + er1 + er2);

  v1[e] = ia * (ea0 * a_vals[e] + ea1 * a_vals[E_EDGES + e] + ea2 * a_vals[2 * E_EDGES + e]);
  v2[e] = ir * (er0 * r_vals[e] + er1 * r_vals[E_EDGES + e] + er2 * r_vals[2 * E_EDGES + e]);
}

// ---------------------------------------------------------------------------
// Kernel 2/4: dense GEMM  C[M x 256] = act(A[M x K]) @ B[K x 256] + bias
// V_WMMA_F32_16X16X4_F32, wave computes a 16x32 tile (A fragment reused
// across two WMMAs). B tile (K x 32) staged in LDS as K-pair float2 rows so
// each B fragment is one bank-conflict-free ds_load_b64.
//
// Fragment layouts (ISA 7.12.2, 32-bit):
//   A 16x4 : a.x = A[lane&15][k0 + 2*(lane>>4)],      a.y = next K
//   B 4x16 : b.x = B[k0 + 2*(lane>>4)][n],            b.y = next K
//   C/D    : acc[v] = C[m0 + v + 8*(lane>>4)][n]
// ---------------------------------------------------------------------------
#define GEMM_KMAX    300            // max K (GEMM1)
#define LDSB_STRIDE  48             // float2 per row; 96 words = 32 mod 64 banks

template <bool LRELU>
__global__ __launch_bounds__(128) void wmma_gemm_kernel(
    const float* __restrict__ A, const float* __restrict__ B,
    const float* __restrict__ bias, float* __restrict__ C,
    int M, int K) {
  __shared__ v2f ldsB[(GEMM_KMAX / 2) * LDSB_STRIDE];   // 57.6 KB

  const int tid  = threadIdx.x;
  const int lane = tid & 31;
  const int wave = tid >> 5;
  const int n0   = blockIdx.x * 32;             // 8 N-positions (256 cols)
  const int mTile = blockIdx.y * 4 + wave;      // 4 waves / block along M
  const bool active = (mTile * 16) < M;

  // ---- cooperative stage of B tile: K x 32 as (K/2) rows of paired floats
  const int nPairs = (K >> 1) * 32;
  for (int idx = tid; idx < nPairs; idx += 128) {
    int r = idx >> 5;          // K-pair index
    int c = idx & 31;          // column within tile
    v2f p;
    p.x = B[(size_t)(2 * r)     * 256 + n0 + c];
    p.y = B[(size_t)(2 * r + 1) * 256 + n0 + c];
    ldsB[r * LDSB_STRIDE + c] = p;
  }
  __syncthreads();

  if (active) {
    const int m0   = mTile * 16;
    const int l16  = lane & 15;
    const int half = lane >> 4;                 // 0 or 1
    const int kk   = half << 1;                 // which K pair of the 4

    const float* aRow = A + (size_t)(m0 + l16) * K;

    v8f acc0 = {};
    v8f acc1 = {};
#pragma unroll 4
    for (int k0 = 0; k0 < K; k0 += 4) {
      v2f a = *(const v2f*)(aRow + k0 + kk);    // 8B aligned: K even, kk even
      if (LRELU) {
        a.x = a.x > 0.0f ? a.x : NEG_SLOPE * a.x;
        a.y = a.y > 0.0f ? a.y : NEG_SLOPE * a.y;
      }
      const int r = (k0 >> 1) + half;           // LDS K-pair row for this lane
      v2f b0 = ldsB[r * LDSB_STRIDE + l16];         // N-tile 0
      v2f b1 = ldsB[r * LDSB_STRIDE + 16 + l16];    // N-tile 1
      acc0 = __builtin_amdgcn_wmma_f32_16x16x4_f32(
          false, a, false, b0, (short)0, acc0, false, false);
      acc1 = __builtin_amdgcn_wmma_f32_16x16x4_f32(
          false, a, false, b1, (short)0, acc1, false, false);
    }

    const float bv0  = bias[n0 + l16];
    const float bv1  = bias[n0 + 16 + l16];
    const int  rbase = m0 + (half << 3);        // +8 rows for upper half-wave
    float* c0 = C + n0 + l16;
    float* c1 = C + n0 + 16 + l16;
#pragma unroll
    for (int v = 0; v < 8; ++v) {
      c0[(size_t)(rbase + v) * 256] = acc0[v] + bv0;
      c1[(size_t)(rbase + v) * 256] = acc1[v] + bv1;
    }
  }
}

// ---------------------------------------------------------------------------
// Kernel 3/5: SpMM scatter  H[rows[e], :] += v[e] * S[cols[e], :]
// 64 threads per edge, 4 channels each (float4 load + 4 f32 atomics, L2-res).
// ---------------------------------------------------------------------------
__global__ __launch_bounds__(256) void spmm_scatter_kernel(
    const float* __restrict__ v, const int* __restrict__ rows,
    const int* __restrict__ cols, const float* __restrict__ S,
    float* __restrict__ H) {
  long tid = (long)blockIdx.x * blockDim.x + threadIdx.x;
  int e = (int)(tid >> 6);
  if (e >= E_EDGES) return;
  int c = ((int)tid & 63) << 2;

  float w = v[e];
  int r   = rows[e];
  int col = cols[e];
  v4f s = *(const v4f*)(S + (size_t)col * 256 + c);
  float* dst = H + (size_t)r * 256 + c;
  atomicAdd(dst + 0, w * s.x);
  atomicAdd(dst + 1, w * s.y);
  atomicAdd(dst + 2, w * s.z);
  atomicAdd(dst + 3, w * s.w);
}

// ---------------------------------------------------------------------------
// Kernel 6: in-place row L2 normalization (one wave per 256-wide row)
// ---------------------------------------------------------------------------
__global__ __launch_bounds__(256) void l2norm_kernel(float* __restrict__ Y) {
  const int lane = threadIdx.x & 31;
  const int wave = threadIdx.x >> 5;
  const int row  = blockIdx.x * 8 + wave;
  if (row >= N_NODES) return;

  float* p = Y + (size_t)row * 256;
  v4f x0 = *(const v4f*)(p + lane * 4);
  v4f x1 = *(const v4f*)(p + 128 + lane * 4);
  float ss = x0.x * x0.x + x0.y * x0.y + x0.z * x0.z + x0.w * x0.w
           + x1.x * x1.x + x1.y * x1.y + x1.z * x1.z + x1.w * x1.w;
#pragma unroll
  for (int off = 16; off > 0; off >>= 1)
    ss += __shfl_xor(ss, off, 32);

  float inv = 1.0f / fmaxf(sqrtf(ss), L2_EPS);
  x0 *= inv;
  x1 *= inv;
  *(v4f*)(p + lane * 4)       = x0;
  *(v4f*)(p + 128 + lane * 4) = x1;
}

// ---------------------------------------------------------------------------
// Launch: v1/v2 -> GEMM1 -> SpMM1 -> GEMM2(fused lrelu) -> SpMM2 -> L2 norm
// ---------------------------------------------------------------------------
extern "C" void kernel_launch(void* const* d_in, const int* in_sizes, int n_in,
                              void* d_out, int out_size, void* d_ws, size_t ws_size,
                              hipStream_t stream) {
  const float* x      = (const float*)d_in[0];
  const float* a_vals = (const float*)d_in[1];
  const int*   a_rows = (const int*)  d_in[2];
  const int*   a_cols = (const int*)  d_in[3];
  const float* r_vals = (const float*)d_in[4];
  const int*   r_rows = (const int*)  d_in[5];
  const int*   r_cols = (const int*)  d_in[6];
  const float* w1     = (const float*)d_in[7];
  const float* b1     = (const float*)d_in[8];
  const float* w2     = (const float*)d_in[9];
  const float* b2     = (const float*)d_in[10];
  const float* a_att  = (const float*)d_in[11];
  const float* r_att  = (const float*)d_in[12];

  float* out = (float*)d_out;
  float* ws  = (float*)d_ws;

  float* v1 = ws;                                   // E
  float* v2 = ws + E_EDGES;                         // E
  float* s1 = ws + 2 * (size_t)E_EDGES;             // N*256 (reused as s2)
  float* h  = s1 + (size_t)N_NODES * H_F;           // N*256

  const size_t featBytes = (size_t)N_NODES * H_F * sizeof(float);
  hipMemsetAsync(h,   0, featBytes, stream);
  hipMemsetAsync(out, 0, featBytes, stream);

  // per-edge mixing weights
  edge_weights_kernel<<<(E_EDGES + 255) / 256, 256, 0, stream>>>(
      a_vals, r_vals, a_att, r_att, v1, v2);

  // s1 = x @ w1 + b1       (M=20000, K=300)
  {
    dim3 grid(H_F / 32, (N_NODES / 16 + 3) / 4);
    wmma_gemm_kernel<false><<<grid, 128, 0, stream>>>(x, w1, b1, s1, N_NODES, IN_F);
  }

  // h = scatter(v1 * s1)
  {
    long threads = (long)E_EDGES * 64;
    spmm_scatter_kernel<<<(unsigned)((threads + 255) / 256), 256, 0, stream>>>(
        v1, a_rows, a_cols, s1, h);
  }

  // s2 = lrelu(h) @ w2 + b2   (M=20000, K=256), s2 reuses s1 buffer
  {
    dim3 grid(OUT_F / 32, (N_NODES / 16 + 3) / 4);
    wmma_gemm_kernel<true><<<grid, 128, 0, stream>>>(h, w2, b2, s1, N_NODES, H_F);
  }

  // out = scatter(v2 * s2)
  {
    long threads = (long)E_EDGES * 64;
    spmm_scatter_kernel<<<(unsigned)((threads + 255) / 256), 256, 0, stream>>>(
        v2, r_rows, r_cols, s1, out);
  }

  // row-wise L2 normalize in place
  l2norm_kernel<<<(N_NODES + 7) / 8, 256, 0, stream>>>(out);
}